// ScatterObservedPorts_39582418600447
// MI455X (gfx1250) — compile-verified
//
#include <hip/hip_runtime.h>

// ---------------------------------------------------------------------------
// ScatterObservedPorts: out[b, n, 0] = sum_{i: idx[i]==n} inputs[b, i]
// (indices unique -> columnwise scatter into zeros).
// Memory-bound: 409.6 MB streamed out -> ~17.6 us floor at 23.3 TB/s.
// Single pass: out[b,n] = inv[n] >= 0 ? in[b, inv[n]] : 0, with inv built
// once in d_ws. Output written with coalesced non-temporal b128 stores so the
// 410 MB stream doesn't evict the L2-resident inputs (4 MB) / inv (200 KB).
// ---------------------------------------------------------------------------

#define NUM_NODES     50000
#define NUM_OBSERVED  512
#define BATCH         2048

#define THREADS          256          // 8 waves (wave32) per block
#define COLS_PER_THREAD  4            // one b128 NT store per row per thread
#define COLS_PER_BLOCK   (THREADS * COLS_PER_THREAD)   // 1024 columns
#define ROWS_PER_BLOCK   16

typedef float v4f __attribute__((ext_vector_type(4)));
typedef int   v4i __attribute__((ext_vector_type(4)));

// --- Pass 1a: inv[n] = -1 ----------------------------------------------------
__global__ void sop_init_inv(int* __restrict__ inv) {
    int n = blockIdx.x * blockDim.x + threadIdx.x;
    if (n < NUM_NODES) inv[n] = -1;
}

// --- Pass 1b: inv[idx[i]] = i ------------------------------------------------
__global__ void sop_scatter_inv(const int* __restrict__ idx, int* __restrict__ inv) {
    int i = blockIdx.x * blockDim.x + threadIdx.x;
    if (i < NUM_OBSERVED) {
        int n = idx[i];
        if (n >= 0 && n < NUM_NODES) inv[n] = i;
    }
}

// --- Pass 2: stream the output -----------------------------------------------
// Block (bx, by): columns [bx*1024, +1024), rows [by*16, +16). Each thread
// loads its 4 inv entries with ONE global_load_b128, hoists the 4 predicates,
// then emits 16 coalesced non-temporal global_store_b128 (one per row).
// Index stride 97 > 4 => at most one gather load per 4-column group per row,
// and it hits L2 (inputs are only 4 MB).
__global__ void __launch_bounds__(THREADS)
sop_scatter_rows(const float* __restrict__ in,
                 const int*   __restrict__ inv,
                 float*       __restrict__ out) {
    const int col0 = blockIdx.x * COLS_PER_BLOCK + threadIdx.x * COLS_PER_THREAD;
    if (col0 >= NUM_NODES) return;   // NUM_NODES % 4 == 0 -> group all in/out

    // One b128 load of this thread's 4 column->input mappings.
    const v4i iv = *(const v4i*)(inv + col0);
    const int i0 = iv.x, i1 = iv.y, i2 = iv.z, i3 = iv.w;
    const bool m0 = (i0 >= 0), m1 = (i1 >= 0), m2 = (i2 >= 0), m3 = (i3 >= 0);

    const int rowBase = blockIdx.y * ROWS_PER_BLOCK;
    const float* __restrict__ inr = in  + (size_t)rowBase * NUM_OBSERVED;
    float*       __restrict__ orw = out + (size_t)rowBase * NUM_NODES + col0;

#pragma unroll
    for (int r = 0; r < ROWS_PER_BLOCK; ++r) {
        v4f v = (v4f)(0.0f);
        if (m0) v.x = inr[i0];        // exec-predicated L2-hit gathers
        if (m1) v.y = inr[i1];
        if (m2) v.z = inr[i2];
        if (m3) v.w = inr[i3];

        // global_store_b128 th:TH_NT on gfx1250
        __builtin_nontemporal_store(v, (v4f*)orw);

        inr += NUM_OBSERVED;
        orw += NUM_NODES;
    }
}

extern "C" void kernel_launch(void* const* d_in, const int* in_sizes, int n_in,
                              void* d_out, int out_size, void* d_ws, size_t ws_size,
                              hipStream_t stream) {
    const float* inputs = (const float*)d_in[0];  // [BATCH, NUM_OBSERVED] f32
    const int*   idx    = (const int*)d_in[1];    // [NUM_OBSERVED] int
    float*       out    = (float*)d_out;          // [BATCH, NUM_NODES, 1] f32
    int*         inv    = (int*)d_ws;             // 50000 ints = 200 KB scratch

    (void)in_sizes; (void)n_in; (void)out_size; (void)ws_size;

    // Build inverse column map (kernels serialize on `stream`).
    sop_init_inv<<<(NUM_NODES + THREADS - 1) / THREADS, THREADS, 0, stream>>>(inv);
    sop_scatter_inv<<<(NUM_OBSERVED + THREADS - 1) / THREADS, THREADS, 0, stream>>>(idx, inv);

    // Streaming scatter: 49 column tiles x 128 row tiles = 6272 blocks.
    dim3 grid((NUM_NODES + COLS_PER_BLOCK - 1) / COLS_PER_BLOCK,
              BATCH / ROWS_PER_BLOCK);
    sop_scatter_rows<<<grid, THREADS, 0, stream>>>(inputs, inv, out);
}